// LengthRegulatorWithAlignment_23974507446349
// MI455X (gfx1250) — compile-verified
//
#include <hip/hip_runtime.h>
#include <hip/hip_bf16.h>
#include <stdint.h>

// Problem constants from the reference: B=32, P=512, D=384, max_len=2048.
#define P_PH   512
#define D_DIM  384
#define T_MAX  2048
#define TILE_T 64
#define VEC_PER_ROW (D_DIM / 4)   // 96 float4 per row

typedef float v4f __attribute__((ext_vector_type(4)));

// ---------------------------------------------------------------------------
// Kernel 1: per-batch inclusive scan of durations -> ends[b,p]; mel_lens out.
// One block per batch, 512 threads (16 wave32 waves), Hillis-Steele in LDS.
// ---------------------------------------------------------------------------
__global__ __launch_bounds__(P_PH)
void lr_scan_kernel(const int* __restrict__ dur,
                    int* __restrict__ ends_ws,
                    int* __restrict__ mel_lens)
{
    __shared__ int s[P_PH];
    const int b   = blockIdx.x;
    const int tid = threadIdx.x;

    s[tid] = dur[b * P_PH + tid];
    __syncthreads();

    #pragma unroll
    for (int off = 1; off < P_PH; off <<= 1) {
        int add = (tid >= off) ? s[tid - off] : 0;
        __syncthreads();
        s[tid] += add;
        __syncthreads();
    }

    ends_ws[b * P_PH + tid] = s[tid];
    if (tid == P_PH - 1) mel_lens[b] = s[tid];   // = ends[:, -1]
}

// ---------------------------------------------------------------------------
// Kernel 2: expand. Each block = one batch x 64-frame tile.
// Stage ends[b,0:512] to LDS via CDNA5 async global->LDS DMA, binary-search
// the source row per frame, then stream coalesced float4 rows with NT stores.
// ---------------------------------------------------------------------------
__global__ __launch_bounds__(256)
void lr_gather_kernel(const float* __restrict__ x,
                      const int*  __restrict__ ends_ws,
                      float* __restrict__ out)
{
    __shared__ __attribute__((aligned(16))) int s_ends[P_PH];
    __shared__ int s_row[TILE_T];

    const int tid = threadIdx.x;
    const int b   = blockIdx.y;
    const int t0  = blockIdx.x * TILE_T;

    // --- Async copy of ends[b, 0:512] (2048 B) into LDS: 256 lanes x 8 B ---
    {
        uint64_t gaddr = (uint64_t)(uintptr_t)(ends_ws + b * P_PH) + (uint64_t)tid * 8u;
        uint32_t laddr = (uint32_t)(uintptr_t)(&s_ends[0]) + (uint32_t)tid * 8u;
        asm volatile("global_load_async_to_lds_b64 %0, %1, off"
                     :: "v"(laddr), "v"(gaddr) : "memory");
        asm volatile("s_wait_asynccnt 0x0" ::: "memory");
    }
    __syncthreads();

    // --- searchsorted(ends, t, 'right') for the 64 frames of this tile ---
    if (tid < TILE_T) {
        const int t = t0 + tid;
        int lo = 0, hi = P_PH;
        #pragma unroll
        for (int it = 0; it < 9; ++it) {       // log2(512) fixed-trip search
            int mid = (lo + hi) >> 1;
            if (s_ends[mid] <= t) lo = mid + 1; else hi = mid;
        }
        int idx = lo < (P_PH - 1) ? lo : (P_PH - 1);   // clip to [0, P-1]
        const int mel = s_ends[P_PH - 1];
        s_row[tid] = (t < mel) ? idx : -1;             // -1 => write zeros
    }
    __syncthreads();

    // --- Stream 64 rows x 96 float4: coalesced loads (L2-hot) + NT stores ---
    const float* xb = x   + (size_t)b * P_PH * D_DIM;
    float*       ob = out + ((size_t)b * T_MAX + t0) * D_DIM;

    const int total = TILE_T * VEC_PER_ROW;            // 6144 float4
    #pragma unroll 4
    for (int e = tid; e < total; e += 256) {
        const int tl  = e / VEC_PER_ROW;               // frame within tile
        const int c   = e - tl * VEC_PER_ROW;          // float4 column
        const int row = s_row[tl];
        v4f val = (v4f)(0.0f);
        if (row >= 0) {
            val = *(const v4f*)(xb + (size_t)row * D_DIM + (size_t)c * 4);
        }
        __builtin_nontemporal_store(val, (v4f*)(ob + (size_t)tl * D_DIM + (size_t)c * 4));
    }
}

// ---------------------------------------------------------------------------
extern "C" void kernel_launch(void* const* d_in, const int* in_sizes, int n_in,
                              void* d_out, int out_size, void* d_ws, size_t ws_size,
                              hipStream_t stream)
{
    const float* x   = (const float*)d_in[0];   // (B, 512, 384) f32
    const int*   dur = (const int*)  d_in[1];   // (B, 512) i32
    // d_in[2] = max_len scalar (2048, baked into T_MAX)

    const int B = in_sizes[1] / P_PH;

    int*   ends_ws = (int*)d_ws;                                   // B*512 ints
    float* out     = (float*)d_out;                                // (B,2048,384) f32
    int*   mel     = (int*)d_out + (size_t)B * T_MAX * D_DIM;      // (B,) i32 tail

    lr_scan_kernel<<<dim3(B), dim3(P_PH), 0, stream>>>(dur, ends_ws, mel);

    dim3 grid(T_MAX / TILE_T, B);
    lr_gather_kernel<<<grid, dim3(256), 0, stream>>>(x, ends_ws, out);
}